// PolarNet_74947179315736
// MI455X (gfx1250) — compile-verified
//
#include <hip/hip_runtime.h>
#include <hip/hip_bf16.h>
#include <stdint.h>

typedef __attribute__((ext_vector_type(2))) float v2f;
typedef __attribute__((ext_vector_type(8))) float v8f;
typedef __attribute__((ext_vector_type(4))) unsigned int u32x4;
typedef __attribute__((ext_vector_type(8))) int i32x8;
typedef __attribute__((ext_vector_type(4))) int i32x4;

#define B_  16
#define C_  256
#define H_  80
#define W_  80
#define N_  (H_ * W_)
#define EPSV 1e-5f

#if __has_builtin(__builtin_amdgcn_tensor_load_to_lds) && __has_builtin(__builtin_amdgcn_s_wait_tensorcnt)
#define USE_TDM 1
#else
#define USE_TDM 0
#endif

// ---------------------------------------------------------------------------
// Kernel 1: fused QKV GEMM via V_WMMA_F32_16X16X4_F32.
// Block = one (batch, 16-wide n-tile); 8 waves x 3 tile-pairs (2 independent
// accumulator chains per wave, shared B operand). x tile (256x16) lands in
// LDS via the Tensor Data Mover (TENSORcnt), falling back to a cooperative
// load if the builtin is unavailable. Outputs q,k,v stored [b][c_out][n].
// ---------------------------------------------------------------------------
__global__ __launch_bounds__(256) void qkv_gemm_wmma(
    const float* __restrict__ x,
    const float* __restrict__ Wq, const float* __restrict__ bq,
    const float* __restrict__ Wk, const float* __restrict__ bk,
    const float* __restrict__ Wv, const float* __restrict__ bv,
    float* __restrict__ q, float* __restrict__ k, float* __restrict__ v)
{
    __shared__ float tile[C_ * 16];          // [c][nn], 16 KB

    const int b   = blockIdx.y;
    const int n0  = blockIdx.x * 16;
    const int tid = threadIdx.x;

#if USE_TDM
    // --- Tensor Data Mover: 2D tile (tile_dim0=16 cols, tile_dim1=256 rows,
    //     row stride = N_) from global x[b][*][n0..n0+15] into LDS [c][16].
    if (tid < 32) {                          // one TDM issue per block (wave 0)
        const uint64_t ga  = (uint64_t)(uintptr_t)(x + (size_t)b * C_ * N_ + n0);
        const uint32_t lds = (uint32_t)(uintptr_t)tile;   // flat低32 = LDS offset

        u32x4 g0 = {
            1u,                                            // count=1, load desc
            lds,                                           // lds_addr
            (uint32_t)ga,                                  // global_addr[31:0]
            ((uint32_t)(ga >> 32) & 0x01FFFFFFu) | 0x80000000u // ga[56:32] | type=2
        };
        i32x8 g1 = {
            (int)0x00020000u,                              // data_size = 4B
            (int)(((uint32_t)N_ & 0xFFFFu) << 16),         // tensor_dim0[15:0]
            (int)(((uint32_t)N_ >> 16) | ((uint32_t)C_ << 16)), // dim0[31:16] | dim1[15:0]
            (int)(16u << 16),                              // tile_dim0 = 16
            (int)C_,                                       // tile_dim1 = 256
            (int)N_,                                       // tensor_dim0_stride
            0, 0
        };
        i32x4 z4 = {0, 0, 0, 0};
#if defined(__clang_major__) && (__clang_major__ >= 23)
        i32x8 z8 = {0, 0, 0, 0, 0, 0, 0, 0};
        __builtin_amdgcn_tensor_load_to_lds(g0, g1, z4, z4, z8, 0);
#else
        __builtin_amdgcn_tensor_load_to_lds(g0, g1, z4, z4, 0);
#endif
        __builtin_amdgcn_s_wait_tensorcnt(0);
    }
    __syncthreads();
#else
    {   // Cooperative fallback load: 256x16 x-tile -> LDS.
        const float* xs = x + (size_t)b * C_ * N_ + n0;
        const int nn    = tid & 15;
        const int cbase = tid >> 4;
        #pragma unroll
        for (int i = 0; i < 16; ++i) {
            const int c = cbase + i * 16;
            tile[c * 16 + nn] = xs[(size_t)c * N_ + nn];
        }
    }
    __syncthreads();
#endif

    const int wave = tid >> 5;               // 0..7
    const int lane = tid & 31;
    const int lm   = lane & 15;
    const int half = lane >> 4;              // 0 or 1

    #pragma unroll
    for (int t = 0; t < 6; t += 2) {
        const int T0 = wave * 6 + t;         // 0..47
        const int T1 = T0 + 1;
        const int mat0 = T0 >> 4, m00 = (T0 & 15) * 16;
        const int mat1 = T1 >> 4, m01 = (T1 & 15) * 16;

        const float* Wm0 = (mat0 == 0) ? Wq : ((mat0 == 1) ? Wk : Wv);
        const float* bm0 = (mat0 == 0) ? bq : ((mat0 == 1) ? bk : bv);
        float*       om0 = (mat0 == 0) ? q  : ((mat0 == 1) ? k  : v);
        const float* Wm1 = (mat1 == 0) ? Wq : ((mat1 == 1) ? Wk : Wv);
        const float* bm1 = (mat1 == 0) ? bq : ((mat1 == 1) ? bk : bv);
        float*       om1 = (mat1 == 0) ? q  : ((mat1 == 1) ? k  : v);

        const v2f* ap0 = (const v2f*)(Wm0 + (size_t)(m00 + lm) * C_);
        const v2f* ap1 = (const v2f*)(Wm1 + (size_t)(m01 + lm) * C_);

        v8f acc0 = {};
        v8f acc1 = {};
        #pragma unroll 4
        for (int k0 = 0; k0 < C_; k0 += 4) {
            // A (16x4): lanes 0-15 K={k0,k0+1}, lanes 16-31 K={k0+2,k0+3}
            v2f a0 = ap0[(k0 >> 1) + half];
            v2f a1 = ap1[(k0 >> 1) + half];
            // B (4x16): shared by both tiles of the pair
            v2f bb;
            bb.x = tile[(k0 + 2 * half)     * 16 + lm];
            bb.y = tile[(k0 + 2 * half + 1) * 16 + lm];
            acc0 = __builtin_amdgcn_wmma_f32_16x16x4_f32(
                       false, a0, false, bb, (short)0, acc0, false, false);
            acc1 = __builtin_amdgcn_wmma_f32_16x16x4_f32(
                       false, a1, false, bb, (short)0, acc1, false, false);
        }

        // D (16x16): VGPR r -> M = r + 8*half, N = lm.
        float* ob0 = om0 + (size_t)b * C_ * N_ + n0 + lm;
        float* ob1 = om1 + (size_t)b * C_ * N_ + n0 + lm;
        #pragma unroll
        for (int r = 0; r < 8; ++r) {
            const int r0 = m00 + half * 8 + r;
            const int r1 = m01 + half * 8 + r;
            ob0[(size_t)r0 * N_] = acc0[r] + bm0[r0];
            ob1[(size_t)r1 * N_] = acc1[r] + bm1[r1];
        }
    }
}

// ---------------------------------------------------------------------------
// Kernel 2: scores[b][j][n] = sum_c q[b][c][n] * k[b][c][nbr(n,j)], plus the
// (center, others_mean) output. Thread per pixel.
// ---------------------------------------------------------------------------
__global__ __launch_bounds__(256) void scores_kernel(
    const float* __restrict__ q, const float* __restrict__ k,
    float* __restrict__ scores, float* __restrict__ out01)
{
    const int b = blockIdx.y;
    const int n = blockIdx.x * 256 + threadIdx.x;
    const int r = n / W_;
    const int c = n % W_;

    int nbr[9];
    #pragma unroll
    for (int j = 0; j < 9; ++j) {
        int ri = r + (j / 3) - 1; ri = ri < 0 ? 0 : (ri > H_ - 1 ? H_ - 1 : ri);
        int ci = c + (j % 3) - 1; ci = ci < 0 ? 0 : (ci > W_ - 1 ? W_ - 1 : ci);
        nbr[j] = ri * W_ + ci;
    }

    float s[9] = {};
    const float* qp = q + (size_t)b * C_ * N_ + n;
    const float* kp = k + (size_t)b * C_ * N_;
    #pragma unroll 2
    for (int ch = 0; ch < C_; ++ch) {
        const float  qv = qp[(size_t)ch * N_];
        const float* kr = kp + (size_t)ch * N_;
        #pragma unroll
        for (int j = 0; j < 9; ++j) s[j] = fmaf(qv, kr[nbr[j]], s[j]);
    }

    float tot = 0.0f;
    #pragma unroll
    for (int j = 0; j < 9; ++j) {
        scores[((size_t)b * 9 + j) * N_ + n] = s[j];
        tot += s[j];
    }
    out01[((size_t)b * N_ + n) * 2 + 0] = s[4];
    out01[((size_t)b * N_ + n) * 2 + 1] = (tot - s[4]) * 0.125f;
}

// ---------------------------------------------------------------------------
// Kernel 3: xr = x + scores . v_nbr, then spatial LayerNorm per (b,c).
// One block per (b,c); 25 pixels/thread kept in registers, LDS tree
// reduction for mean/var. Deterministic (no atomics).
// ---------------------------------------------------------------------------
__global__ __launch_bounds__(256) void attnv_layernorm(
    const float* __restrict__ x, const float* __restrict__ v,
    const float* __restrict__ scores, float* __restrict__ xout)
{
    const int bc  = blockIdx.x;
    const int b   = bc >> 8;
    const int ch  = bc & 255;
    const int tid = threadIdx.x;

    const float* xp = x      + (size_t)b * C_ * N_ + (size_t)ch * N_;
    const float* vp = v      + (size_t)b * C_ * N_ + (size_t)ch * N_;
    const float* sp = scores + (size_t)b * 9 * N_;

    float xr[25];
    float sum = 0.0f, sq = 0.0f;

    #pragma unroll 1
    for (int i = 0; i < 25; ++i) {
        const int n = i * 256 + tid;
        const int r = n / W_;
        const int c = n % W_;
        float nv = 0.0f;
        #pragma unroll
        for (int j = 0; j < 9; ++j) {
            int ri = r + (j / 3) - 1; ri = ri < 0 ? 0 : (ri > H_ - 1 ? H_ - 1 : ri);
            int ci = c + (j % 3) - 1; ci = ci < 0 ? 0 : (ci > W_ - 1 ? W_ - 1 : ci);
            nv = fmaf(sp[(size_t)j * N_ + n], vp[ri * W_ + ci], nv);
        }
        const float val = xp[n] + nv;
        xr[i] = val;
        sum  += val;
        sq    = fmaf(val, val, sq);
    }

    __shared__ float ls[256];
    __shared__ float lq[256];
    ls[tid] = sum; lq[tid] = sq;
    __syncthreads();
    #pragma unroll
    for (int off = 128; off > 0; off >>= 1) {
        if (tid < off) { ls[tid] += ls[tid + off]; lq[tid] += lq[tid + off]; }
        __syncthreads();
    }
    const float mu  = ls[0] * (1.0f / N_);
    const float var = lq[0] * (1.0f / N_) - mu * mu;
    const float rs  = rsqrtf(var + EPSV);

    float* op = xout + (size_t)b * C_ * N_ + (size_t)ch * N_;
    #pragma unroll 1
    for (int i = 0; i < 25; ++i)
        op[i * 256 + tid] = (xr[i] - mu) * rs;
}

// ---------------------------------------------------------------------------
extern "C" void kernel_launch(void* const* d_in, const int* in_sizes, int n_in,
                              void* d_out, int out_size, void* d_ws, size_t ws_size,
                              hipStream_t stream)
{
    const float* x  = (const float*)d_in[0];
    const float* Wq = (const float*)d_in[1];
    const float* bq = (const float*)d_in[2];
    const float* Wk = (const float*)d_in[3];
    const float* bk = (const float*)d_in[4];
    const float* Wv = (const float*)d_in[5];
    const float* bv = (const float*)d_in[6];

    const size_t CN = (size_t)B_ * C_ * N_;   // 26,214,400 floats
    float* ws = (float*)d_ws;
    float* q  = ws;
    float* k  = q + CN;
    float* v  = k + CN;
    float* sc = v + CN;                        // B*9*N floats

    float* out01 = (float*)d_out;                          // (B,H,W,2)
    float* xout  = out01 + (size_t)B_ * N_ * 2;            // (B,C,H,W)

    qkv_gemm_wmma<<<dim3(N_ / 16, B_), 256, 0, stream>>>(
        x, Wq, bq, Wk, bk, Wv, bv, q, k, v);

    scores_kernel<<<dim3(N_ / 256, B_), 256, 0, stream>>>(q, k, sc, out01);

    attnv_layernorm<<<dim3(B_ * C_), 256, 0, stream>>>(x, v, sc, xout);
}